// HodgkinHuxley_44384192037149
// MI455X (gfx1250) — compile-verified
//
#include <hip/hip_runtime.h>

#define HH_DT      0.1f
#define HH_REST_CA 0.363f
#define HH_PF      24   // prefetch distance in timesteps

// sigmoid(x) = 1/(1+exp(-x)) via v_exp_f32 (base-2) + v_rcp_f32.
__device__ __forceinline__ float fast_sigmoid(float x) {
    float e = __builtin_amdgcn_exp2f(x * -1.44269504088896340736f); // exp(-x)
    return __builtin_amdgcn_rcpf(1.0f + e);
}

// Shared simulation body. TT/NN == 0 means "use runtime rt/rn".
template <int TT, int NN>
__device__ __forceinline__ void hh_body(
    const float* __restrict__ i_inj, const float* __restrict__ params,
    const float* __restrict__ init, float* __restrict__ out, int rt, int rn)
{
    const int T = TT ? TT : rt;
    const int N = NN ? NN : rn;

    const int n = blockIdx.x * blockDim.x + threadIdx.x;
    if (n >= N) return;

    // ---- hoist per-neuron parameters into registers (28 coalesced loads) ----
    float P[28];
#pragma unroll
    for (int r = 0; r < 28; ++r) P[r] = params[(size_t)r * N + n];

    const float p_mdp = P[0],  p_is = 1.0f / P[1];
    const float q_mdp = P[3],  q_is = 1.0f / P[4];
    const float n_mdp = P[6],  n_is = 1.0f / P[7];
    const float e_mdp = P[9],  e_is = 1.0f / P[10];
    const float f_mdp = P[12], f_is = 1.0f / P[13];
    const float h_mdp = P[15], h_is = 1.0f / P[16], h_alpha = P[17];
    const float g_ca = P[18], e_ca = P[19];
    const float g_kf = P[20], e_k  = P[21];
    const float g_ks = P[22];
    const float g_l  = P[23], e_l  = P[24];
    const float dt_over_cm = HH_DT / P[25];
    const float rho_dt = P[26] * HH_DT;          // rho_ca * DT
    const float dec_dt = HH_DT / P[27];          // DT / decay_ca

    // gate recurrence folded to one FMA: x' = A*x + B*sigmoid((V-mdp)/scale)
    const float pA = P[2]  / (P[2]  + HH_DT), pB = HH_DT / (P[2]  + HH_DT);
    const float qA = P[5]  / (P[5]  + HH_DT), qB = HH_DT / (P[5]  + HH_DT);
    const float nA = P[8]  / (P[8]  + HH_DT), nB = HH_DT / (P[8]  + HH_DT);
    const float eA = P[11] / (P[11] + HH_DT), eB = HH_DT / (P[11] + HH_DT);
    const float fA = P[14] / (P[14] + HH_DT), fB = HH_DT / (P[14] + HH_DT);
    const float hA = 1.0f - h_alpha;             // h = hA + h_alpha*sigmoid(...)

    // ---- initial state ----
    float V   = init[0 * (size_t)N + n];
    float p   = init[1 * (size_t)N + n];
    float q   = init[2 * (size_t)N + n];
    float g_n = init[3 * (size_t)N + n];
    float e_g = init[4 * (size_t)N + n];
    float f_g = init[5 * (size_t)N + n];
    float cac = init[6 * (size_t)N + n];

    const float* ii = i_inj + n;   // bumped by N each step
    float*       ob = out   + n;   // bumped by 7N each step

    auto step = [&](const float cur) {
        // calcium-dependent gating + currents (old V)
        const float h     = hA + h_alpha * fast_sigmoid((cac - h_mdp) * h_is);
        const float geefh = g_ca * e_g * e_g * f_g * h;
        const float i_ca  = geefh * (V - e_ca);
        const float i_ks  = g_ks * g_n * (V - e_k);
        const float p2    = p * p;
        const float i_kf  = g_kf * p2 * p2 * q * (V - e_k);
        const float i_l   = g_l * (V - e_l);

        // membrane update, then Ca update at the *new* V (matches reference)
        V += (cur - i_ca - i_ks - i_kf - i_l) * dt_over_cm;
        const float i_ca2 = geefh * (V - e_ca);
        cac = cac - i_ca2 * rho_dt - (cac - HH_REST_CA) * dec_dt;

        // five independent gate recurrences at new V (ILP across v_exp chains)
        p   = pA * p   + pB * fast_sigmoid((V - p_mdp) * p_is);
        q   = qA * q   + qB * fast_sigmoid((V - q_mdp) * q_is);
        e_g = eA * e_g + eB * fast_sigmoid((V - e_mdp) * e_is);
        f_g = fA * f_g + fB * fast_sigmoid((V - f_mdp) * f_is);
        g_n = nA * g_n + nB * fast_sigmoid((V - n_mdp) * n_is);

        // coalesced NT streaming stores: out[t, k, n], k in stack order.
        // With constexpr N these are 7 immediate-IOFFSET stores off one base.
        __builtin_nontemporal_store(V,   ob + 0 * (size_t)N);
        __builtin_nontemporal_store(p,   ob + 1 * (size_t)N);
        __builtin_nontemporal_store(q,   ob + 2 * (size_t)N);
        __builtin_nontemporal_store(g_n, ob + 3 * (size_t)N);
        __builtin_nontemporal_store(e_g, ob + 4 * (size_t)N);
        __builtin_nontemporal_store(f_g, ob + 5 * (size_t)N);
        __builtin_nontemporal_store(cac, ob + 6 * (size_t)N);
        ii += N;
        ob += 7 * (size_t)N;
    };

    // main loop with prefetch (no per-iteration bound check) ...
    const int tmain = (T > HH_PF) ? (T - HH_PF) : 0;
#pragma unroll 2
    for (int t = 0; t < tmain; ++t) {
        __builtin_prefetch(ii + HH_PF * (size_t)N, 0, 0); // global_prefetch_b8
        step(__builtin_nontemporal_load(ii));
    }
    // ... tail without prefetch
    for (int t = tmain; t < T; ++t)
        step(__builtin_nontemporal_load(ii));
}

// Specialized for the reference problem size: immediate store offsets,
// known trip count, fully strength-reduced addressing.
__global__ __launch_bounds__(256, 1) void hh_sim_fixed(
    const float* __restrict__ i_inj, const float* __restrict__ params,
    const float* __restrict__ init, float* __restrict__ out)
{
    hh_body<1024, 16384>(i_inj, params, init, out, 0, 0);
}

// Generic fallback for any other T/N.
__global__ __launch_bounds__(256, 1) void hh_sim_generic(
    const float* __restrict__ i_inj, const float* __restrict__ params,
    const float* __restrict__ init, float* __restrict__ out, int T, int N)
{
    hh_body<0, 0>(i_inj, params, init, out, T, N);
}

extern "C" void kernel_launch(void* const* d_in, const int* in_sizes, int n_in,
                              void* d_out, int out_size, void* d_ws, size_t ws_size,
                              hipStream_t stream) {
    const float* i_inj  = (const float*)d_in[0];  // [T, N]
    const float* params = (const float*)d_in[1];  // [28, N]
    const float* init   = (const float*)d_in[2];  // [7, N]
    float*       out    = (float*)d_out;          // [T, 7, N]

    const int N = in_sizes[1] / 28;
    const int T = in_sizes[0] / N;

    const int block = 256;                        // 8 wave32s per workgroup
    const int grid  = (N + block - 1) / block;

    if (T == 1024 && N == 16384) {
        hh_sim_fixed<<<grid, block, 0, stream>>>(i_inj, params, init, out);
    } else {
        hh_sim_generic<<<grid, block, 0, stream>>>(i_inj, params, init, out, T, N);
    }
}